// GuidedAnchoringRPN_18691697672261
// MI455X (gfx1250) — compile-verified
//
#include <hip/hip_runtime.h>
#include <math.h>

// ---------------------------------------------------------------------------
// Guided-Anchoring RPN loss, fused single-pass kernel for MI455X (gfx1250).
// Levels padded so each 256-thread block is level-uniform:
//   lvl0: B*P=65536 -> 256 blocks, lvl1: 16384 -> 64, lvl2: 4096 -> 16,
//   lvl3: 1024 -> 4.  Total 340 blocks.
// Per-block 3-way reduction (loc focal sum, shape loss sum, npos) done with
// V_WMMA_F32_16X16X4_F32 as a wave32 f32 all-reduce, then global atomics into
// 12 per-level accumulators in d_ws; a 1-thread finalize applies the per-level
// means / npos normalization and writes the scalar loss.
// ---------------------------------------------------------------------------

typedef float v2f __attribute__((ext_vector_type(2)));
typedef float v8f __attribute__((ext_vector_type(8)));

// Full 32-lane f32 sum using one WMMA:
//   A(16x4): VGPR0 = v  -> A[m,0]=v[m] (lanes 0-15), A[m,2]=v[m+16] (lanes 16-31)
//            VGPR1 = 0  -> A[m,1]=A[m,3]=0
//   B(4x16): all ones  =>  D[m,n] = rowsumA(m) = v[m] + v[m+16]
//   Sum of the 8 D VGPRs: lanes 0-15 hold sum_{m=0..7}, lanes 16-31 sum_{m=8..15}.
//   One xor-shuffle by 16 completes the reduction (result in every lane).
__device__ __forceinline__ float wave_allreduce_wmma(float v) {
    v2f a; a[0] = v;    a[1] = 0.0f;
    v2f b; b[0] = 1.0f; b[1] = 1.0f;
    v8f c = {};
    c = __builtin_amdgcn_wmma_f32_16x16x4_f32(
        /*neg_a=*/false, a, /*neg_b=*/false, b,
        /*c_mod=*/(short)0, c, /*reuse_a=*/false, /*reuse_b=*/false);
    float s = c[0] + c[1] + c[2] + c[3] + c[4] + c[5] + c[6] + c[7];
    s += __shfl_xor(s, 16, 32);
    return s;
}

// smooth-L1 / bounded-IoU penalty: |x|<0.2 ? 0.5*x^2/0.2 : |x|-0.1
__device__ __forceinline__ float bih(float v) {
    float a = fabsf(v);
    return (a < 0.2f) ? (2.5f * v * v) : (a - 0.1f);
}

#define GA_B 4
#define GA_G 24
#define GA_BG 96   // B*G

__global__ void ga_init_kernel(float* acc) {
    if (threadIdx.x < 12) acc[threadIdx.x] = 0.0f;
}

__global__ __launch_bounds__(256)
void ga_main_kernel(const float* __restrict__ gt,
                    const float* __restrict__ lp0, const float* __restrict__ sp0,
                    const float* __restrict__ lp1, const float* __restrict__ sp1,
                    const float* __restrict__ lp2, const float* __restrict__ sp2,
                    const float* __restrict__ lp3, const float* __restrict__ sp3,
                    float* __restrict__ acc)
{
    const int FHs[4]  = {128, 64, 32, 16};
    const int STRs[4] = {8, 16, 32, 64};

    const int blk = blockIdx.x;
    int lvl, base;
    if      (blk < 256) { lvl = 0; base = 0;   }
    else if (blk < 320) { lvl = 1; base = 256; }
    else if (blk < 336) { lvl = 2; base = 320; }
    else                { lvl = 3; base = 336; }

    const int   fh   = FHs[lvl];
    const int   fw   = fh;
    const int   P    = fh * fw;
    const float strf = (float)STRs[lvl];

    const float* __restrict__ lp = (lvl == 0) ? lp0 : (lvl == 1) ? lp1 : (lvl == 2) ? lp2 : lp3;
    const float* __restrict__ sp = (lvl == 0) ? sp0 : (lvl == 1) ? sp1 : (lvl == 2) ? sp2 : sp3;

    const int tid = threadIdx.x;

    // ---- GT preprocessing into LDS (per (b,g): box, area, level flag, center region) ----
    __shared__ float s_x1[GA_BG], s_y1[GA_BG], s_x2[GA_BG], s_y2[GA_BG], s_ar[GA_BG];
    __shared__ int   s_ix1[GA_BG], s_iy1[GA_BG], s_ix2[GA_BG], s_iy2[GA_BG], s_on[GA_BG];
    if (tid < GA_BG) {
        float x1 = gt[4*tid+0], y1 = gt[4*tid+1], x2 = gt[4*tid+2], y2 = gt[4*tid+3];
        float bw = x2 - x1, bh = y2 - y1;
        float cx = 0.5f*(x1 + x2), cy = 0.5f*(y1 + y2);
        s_x1[tid] = x1; s_y1[tid] = y1; s_x2[tid] = x2; s_y2[tid] = y2;
        s_ar[tid] = bw * bh;
        float sa = sqrtf(fmaxf(bw*bh, 1e-6f));
        float lv = floorf(log2f(fmaxf(sa, 1.0f))) - 2.0f;
        lv = fminf(fmaxf(lv, 0.0f), 3.0f);
        s_on[tid] = ((int)lv == lvl) ? 1 : 0;
        const float rr = 0.2f * 0.5f;     // CENTER_RATIO/2
        s_ix1[tid] = (int)fmaxf(0.0f, floorf((cx - bw*rr) / strf));
        s_iy1[tid] = (int)fmaxf(0.0f, floorf((cy - bh*rr) / strf));
        s_ix2[tid] = min(fw, (int)floorf((cx + bw*rr) / strf) + 1);
        s_iy2[tid] = min(fh, (int)floorf((cy + bh*rr) / strf) + 1);
    }
    __syncthreads();

    // ---- this thread's (b, position) ----
    const int idx  = (blk - base) * 256 + tid;   // in [0, B*P)
    const int b    = idx / P;
    const int ppos = idx - b * P;
    const int y    = ppos / fw;
    const int x    = ppos - y * fw;

    const float px = ((float)x + 0.5f) * strf;   // pixel center
    const float py = ((float)y + 0.5f) * strf;

    // ---- loc target: any on-level GT whose center region covers (x,y) ----
    int tgt = 0;
    for (int g = 0; g < GA_G; ++g) {
        int j = b * GA_G + g;
        if (s_on[j] && x >= s_ix1[j] && x < s_ix2[j] && y >= s_iy1[j] && y < s_iy2[j]) tgt = 1;
    }

    // ---- focal loss term ----
    float pr = lp[b * P + ppos];
    float p  = 1.0f / (1.0f + expf(-pr));
    float pt = tgt ? p : (1.0f - p);
    float at = tgt ? 0.25f : 0.75f;
    float om = 1.0f - pt;
    float lloc = -at * om * om * logf(fmaxf(pt, 1e-6f));

    // ---- 9 approx anchors (3 scales x 3 ratios) ----
    float aw[9], ah[9];
    {
        const float SC[3] = {1.0f, 1.2599210498948732f, 1.5874010519681994f}; // 2^(i/3)
        const float SR[3] = {1.4142135623730951f, 1.0f, 0.7071067811865476f}; // 1/sqrt(r), r=0.5,1,2
        #pragma unroll
        for (int i = 0; i < 3; ++i) {
            float s = 8.0f * strf * SC[i];   // OCTAVE_BASE * stride * 2^(i/3)
            #pragma unroll
            for (int j = 0; j < 3; ++j) {
                aw[i*3+j] = s * SR[j];       // w = s / sqrt(r)
                ah[i*3+j] = s / SR[j];       // h = s * sqrt(r)
            }
        }
    }

    // ---- max-over-anchors IoU per GT, argmax over GT (first-wins) ----
    float best = -1.0f; int bg = 0;
    for (int g = 0; g < GA_G; ++g) {
        int j = b * GA_G + g;
        float gx1 = s_x1[j], gy1 = s_y1[j], gx2 = s_x2[j], gy2 = s_y2[j];
        float ag  = s_ar[j];
        float m = 0.0f;
        #pragma unroll
        for (int a = 0; a < 9; ++a) {
            float hw = 0.5f * aw[a], hh = 0.5f * ah[a];
            float lx = fmaxf(px - hw, gx1), ly = fmaxf(py - hh, gy1);
            float rx = fminf(px + hw, gx2), ry = fminf(py + hh, gy2);
            float iw = fmaxf(rx - lx, 0.0f), ih = fmaxf(ry - ly, 0.0f);
            float inter = iw * ih;
            float uni   = aw[a]*ah[a] + ag - inter;
            float iou   = inter / fmaxf(uni, 1e-6f);
            m = fmaxf(m, iou);
        }
        if (m > best) { best = m; bg = g; }
    }

    // ---- shape (bounded-IoU) term; guided boxes share centers => dx=dy=0 ----
    float lshape = 0.0f, lnpos = 0.0f;
    if (best >= 0.5f) {
        lnpos = 1.0f;
        int j = b * GA_G + bg;
        float gw  = fmaxf(s_x2[j] - s_x1[j], 1.0f);
        float gh  = fmaxf(s_y2[j] - s_y1[j], 1.0f);
        float sqs = 8.0f * strf;                     // square-anchor side (= max(side,1))
        float d0  = sp[(b*2 + 0) * P + ppos];
        float d1  = sp[(b*2 + 1) * P + ppos];
        float nwp = sqs * expf(fminf(d0, 4.0f));
        float nhp = sqs * expf(fminf(d1, 4.0f));
        float nwt = sqs * expf(fminf(logf(gw / sqs), 4.0f));
        float nht = sqs * expf(fminf(logf(gh / sqs), 4.0f));
        float pw = fmaxf(nwp, 1.0f), ph = fmaxf(nhp, 1.0f);
        float tw = fmaxf(nwt, 1.0f), th = fmaxf(nht, 1.0f);
        float lw = 1.0f - fminf(pw, tw) / fmaxf(pw, tw);
        float lh = 1.0f - fminf(ph, th) / fmaxf(ph, th);
        lshape = bih(lw) + bih(lh);   // dx,dy terms are exactly 0
    }

    // ---- block reduction: LDS fold to 32 lanes, then WMMA wave all-reduce ----
    __shared__ float red[3][256];
    red[0][tid] = lloc; red[1][tid] = lshape; red[2][tid] = lnpos;
    __syncthreads();
    if (tid < 32) {   // whole wave 0 active -> EXEC all ones for WMMA
        float a0 = 0.0f, a1 = 0.0f, a2 = 0.0f;
        #pragma unroll
        for (int i = 0; i < 8; ++i) {
            a0 += red[0][tid + 32*i];
            a1 += red[1][tid + 32*i];
            a2 += red[2][tid + 32*i];
        }
        a0 = wave_allreduce_wmma(a0);
        a1 = wave_allreduce_wmma(a1);
        a2 = wave_allreduce_wmma(a2);
        if (tid == 0) {
            atomicAdd(&acc[lvl*3 + 0], a0);
            atomicAdd(&acc[lvl*3 + 1], a1);
            atomicAdd(&acc[lvl*3 + 2], a2);
        }
    }
}

__global__ void ga_final_kernel(const float* __restrict__ acc, float* __restrict__ out) {
    if (threadIdx.x == 0 && blockIdx.x == 0) {
        const float CNT[4] = {65536.0f, 16384.0f, 4096.0f, 1024.0f};  // B * fh * fw
        float loc = 0.0f, shp = 0.0f;
        #pragma unroll
        for (int l = 0; l < 4; ++l) {
            loc += acc[l*3 + 0] / CNT[l];
            shp += acc[l*3 + 1] / fmaxf(acc[l*3 + 2] * 4.0f, 1.0f);
        }
        out[0] = 0.25f * loc + 0.25f * shp;
    }
}

extern "C" void kernel_launch(void* const* d_in, const int* in_sizes, int n_in,
                              void* d_out, int out_size, void* d_ws, size_t ws_size,
                              hipStream_t stream) {
    // setup_inputs() dict order: gt, (loc_pred, shape_pred) per level 0..3
    const float* gt  = (const float*)d_in[0];
    const float* lp0 = (const float*)d_in[1];
    const float* sp0 = (const float*)d_in[2];
    const float* lp1 = (const float*)d_in[3];
    const float* sp1 = (const float*)d_in[4];
    const float* lp2 = (const float*)d_in[5];
    const float* sp2 = (const float*)d_in[6];
    const float* lp3 = (const float*)d_in[7];
    const float* sp3 = (const float*)d_in[8];

    float* acc = (float*)d_ws;   // 12 floats: {loc_sum, shape_sum, npos} x 4 levels

    ga_init_kernel<<<1, 32, 0, stream>>>(acc);
    ga_main_kernel<<<340, 256, 0, stream>>>(gt, lp0, sp0, lp1, sp1, lp2, sp2, lp3, sp3, acc);
    ga_final_kernel<<<1, 1, 0, stream>>>(acc, (float*)d_out);
}